// Sampler_28449863369155
// MI455X (gfx1250) — compile-verified
//
#include <hip/hip_runtime.h>
#include <hip/hip_bf16.h>
#include <math.h>

// ---------------------------------------------------------------------------
// Sampler for MI455X (gfx1250, wave32).
//   B=256 rows, V=128000 logits/row. One 512-thread block (16 wave32) per row.
//   Min traffic = 131MB (one logits read) ~5.6us @ 23.3 TB/s HBM. We make 7
//   row passes but the whole tensor (131MB) fits in the 192MB L2, so passes
//   2..7 are L2-resident; HBM traffic stays ~1x.
//   Exact top-64 per row via LDS radix-select (4x 8-bit digits over a
//   monotonic float->uint key). top_k < 64 guarantees the final threshold is
//   determined by the top-64 values + softmax denominator Z.
//   Gumbel noise reproduces JAX threefry2x32 (key(42)) bit-layout exactly.
// ---------------------------------------------------------------------------

#define NTHR 512

__device__ __forceinline__ unsigned rotl32(unsigned x, unsigned d) {
  return (x << d) | (x >> (32u - d));
}

// JAX threefry2x32, key = (0, 42), counts = iota(n) split in halves.
__device__ __forceinline__ unsigned threefry_bits(unsigned idx, unsigned half) {
  unsigned cnt = (idx < half) ? idx : (idx - half);
  unsigned x0 = cnt;
  unsigned x1 = cnt + half;
  const unsigned ks0 = 0u;
  const unsigned ks1 = 42u;
  const unsigned ks2 = 0x1BD11BDAu ^ 0u ^ 42u;
  unsigned ks[3] = {ks0, ks1, ks2};
  x0 += ks[0];
  x1 += ks[1];
  const unsigned rot[2][4] = {{13u, 15u, 26u, 6u}, {17u, 29u, 16u, 24u}};
#pragma unroll
  for (int g = 0; g < 5; ++g) {
    const unsigned* r = rot[g & 1];
#pragma unroll
    for (int j = 0; j < 4; ++j) {
      x0 += x1;
      x1 = rotl32(x1, r[j]);
      x1 ^= x0;
    }
    x0 += ks[(g + 1) % 3];
    x1 += ks[(g + 2) % 3] + (unsigned)(g + 1);
  }
  return (idx < half) ? x0 : x1;
}

__device__ __forceinline__ float gumbel_at(unsigned idx, unsigned half) {
  unsigned bits = threefry_bits(idx, half);
  // JAX uniform: mantissa fill with exponent 1, then scale to [minval, maxval)
  float f = __uint_as_float((bits >> 9) | 0x3F800000u) - 1.0f;
  float u = f * (1.0f - 1e-10f) + 1e-10f;
  u = fmaxf(u, 1e-10f);
  return -logf(-logf(u));
}

// monotonic float -> uint key (total order, -inf maps low; NaN not expected)
__device__ __forceinline__ unsigned f2key(float f) {
  unsigned u = __float_as_uint(f);
  return (u & 0x80000000u) ? ~u : (u | 0x80000000u);
}
__device__ __forceinline__ float key2f(unsigned k) {
  if (k == 0u) return -__builtin_inff();  // artificial key of min-p-masked elems
  unsigned u = (k & 0x80000000u) ? (k & 0x7FFFFFFFu) : ~k;
  return __uint_as_float(u);
}

// parallel suffix-scan over 256-bin LDS histogram; picks the radix digit that
// contains the rank-`rem` element (counting from the top).
__device__ __forceinline__ void select_digit(unsigned* sHist, volatile unsigned* sh_prefix,
                                             volatile unsigned* sh_rem, int shift, int tid) {
  __syncthreads();
  // in-place Hillis-Steele suffix sum: sHist[d] = sum_{j>=d} hist[j]
  for (unsigned off = 1; off < 256; off <<= 1) {
    unsigned add = 0u;
    if (tid < 256 && tid + off < 256) add = sHist[tid + off];
    __syncthreads();
    if (tid < 256) sHist[tid] += add;
    __syncthreads();
  }
  unsigned rem = *sh_rem;
  __syncthreads();
  if (tid < 256) {
    unsigned sd = sHist[tid];
    unsigned sd1 = (tid == 255) ? 0u : sHist[tid + 1];
    if (sd >= rem && sd1 < rem) {  // unique bin (suffix is non-increasing)
      *sh_prefix |= ((unsigned)tid) << shift;
      *sh_rem = rem - sd1;
    }
  }
  __syncthreads();
}

__global__ __launch_bounds__(NTHR) void Sampler_28449863369155_kernel(
    const float* __restrict__ logits, const float* __restrict__ temperature,
    const float* __restrict__ min_p, const int* __restrict__ top_k,
    const float* __restrict__ top_p, float* __restrict__ out, int V, unsigned half) {
  const int b = blockIdx.x;
  const int tid = threadIdx.x;
  const int nvec = V >> 2;
  const float* row = logits + (size_t)b * (size_t)V;
  const float4* rowv = (const float4*)row;
  const float NEG_INF = -__builtin_inff();

  __shared__ float sV[NTHR];
  __shared__ int sI[NTHR];
  __shared__ unsigned sHist[256];
  __shared__ float sTop[64];
  __shared__ unsigned sCnt;
  __shared__ float sh_invT, sh_mScaled, sh_minpTh, sh_Z, sh_thresh;
  __shared__ int sh_greedy;
  __shared__ unsigned sh_prefix, sh_rem;

  // ---------------- Pass 1: row max + argmax (first occurrence) -------------
  {
    float bm = NEG_INF;
    int bi = 0x7FFFFFFF;
    for (int i4 = tid; i4 < nvec; i4 += NTHR) {
      __builtin_prefetch(rowv + i4 + 4 * NTHR, 0, 0);  // global_prefetch_b8
      float4 v = rowv[i4];
      int base = i4 << 2;
      if (v.x > bm || (v.x == bm && base + 0 < bi)) { bm = v.x; bi = base + 0; }
      if (v.y > bm || (v.y == bm && base + 1 < bi)) { bm = v.y; bi = base + 1; }
      if (v.z > bm || (v.z == bm && base + 2 < bi)) { bm = v.z; bi = base + 2; }
      if (v.w > bm || (v.w == bm && base + 3 < bi)) { bm = v.w; bi = base + 3; }
    }
    sV[tid] = bm;
    sI[tid] = bi;
    __syncthreads();
    for (int s = NTHR / 2; s > 0; s >>= 1) {
      if (tid < s) {
        float v2 = sV[tid + s];
        int i2 = sI[tid + s];
        if (v2 > sV[tid] || (v2 == sV[tid] && i2 < sI[tid])) { sV[tid] = v2; sI[tid] = i2; }
      }
      __syncthreads();
    }
    if (tid == 0) {
      float m = sV[0];
      sh_greedy = sI[0];
      float t = temperature[b];
      float st = (t < 1e-5f) ? 1.0f : t;
      float invT = 1.0f / st;
      sh_invT = invT;
      float ms = m * invT;  // max of scaled logits (same argmax: invT > 0)
      sh_mScaled = ms;
      float mp = min_p[b];
      // keep iff exp(s - ms) >= min_p  <=>  s >= ms + log(min_p)
      sh_minpTh = (mp > 0.0f) ? (ms + logf(mp)) : NEG_INF;
      sh_prefix = 0u;
      sh_rem = 64u;  // rank of the 64th-largest (descending)
    }
    __syncthreads();
  }
  const float invT = sh_invT;
  const float mS = sh_mScaled;
  const float minpTh = sh_minpTh;

  // ---------------- Pass 2: softmax denom Z + radix digit 0 (bits 31:24) ----
  {
    if (tid < 256) sHist[tid] = 0u;
    __syncthreads();
    float z = 0.0f;
    for (int i4 = tid; i4 < nvec; i4 += NTHR) {
      float4 v = rowv[i4];
      float s;
      unsigned key;
      s = v.x * invT;
      if (s >= minpTh) { z += expf(s - mS); key = f2key(s); } else key = 0u;
      atomicAdd(&sHist[key >> 24], 1u);
      s = v.y * invT;
      if (s >= minpTh) { z += expf(s - mS); key = f2key(s); } else key = 0u;
      atomicAdd(&sHist[key >> 24], 1u);
      s = v.z * invT;
      if (s >= minpTh) { z += expf(s - mS); key = f2key(s); } else key = 0u;
      atomicAdd(&sHist[key >> 24], 1u);
      s = v.w * invT;
      if (s >= minpTh) { z += expf(s - mS); key = f2key(s); } else key = 0u;
      atomicAdd(&sHist[key >> 24], 1u);
    }
    __syncthreads();
    select_digit(sHist, &sh_prefix, &sh_rem, 24, tid);
    // Z reduction (reuse sV)
    sV[tid] = z;
    __syncthreads();
    for (int s = NTHR / 2; s > 0; s >>= 1) {
      if (tid < s) sV[tid] += sV[tid + s];
      __syncthreads();
    }
    if (tid == 0) sh_Z = sV[0];
    __syncthreads();
  }

  // ---------------- Passes 3-5: radix digits 1..3 ---------------------------
  for (int shift = 16; shift >= 0; shift -= 8) {
    unsigned prefix = sh_prefix;
    const int hs = shift + 8;  // 24,16,8 (safe shifts)
    if (tid < 256) sHist[tid] = 0u;
    __syncthreads();
    for (int i4 = tid; i4 < nvec; i4 += NTHR) {
      float4 v = rowv[i4];
      float s;
      unsigned key;
      s = v.x * invT;
      key = (s >= minpTh) ? f2key(s) : 0u;
      if ((key >> hs) == (prefix >> hs)) atomicAdd(&sHist[(key >> shift) & 255u], 1u);
      s = v.y * invT;
      key = (s >= minpTh) ? f2key(s) : 0u;
      if ((key >> hs) == (prefix >> hs)) atomicAdd(&sHist[(key >> shift) & 255u], 1u);
      s = v.z * invT;
      key = (s >= minpTh) ? f2key(s) : 0u;
      if ((key >> hs) == (prefix >> hs)) atomicAdd(&sHist[(key >> shift) & 255u], 1u);
      s = v.w * invT;
      key = (s >= minpTh) ? f2key(s) : 0u;
      if ((key >> hs) == (prefix >> hs)) atomicAdd(&sHist[(key >> shift) & 255u], 1u);
    }
    __syncthreads();
    select_digit(sHist, &sh_prefix, &sh_rem, shift, tid);
  }

  // ---------------- Pass 6: collect strictly-greater-than-64th, thresholds --
  {
    if (tid == 0) sCnt = 0u;
    __syncthreads();
    const unsigned x64key = sh_prefix;  // exact key of 64th-largest masked value
    for (int i4 = tid; i4 < nvec; i4 += NTHR) {
      float4 v = rowv[i4];
      float s;
      unsigned key;
      s = v.x * invT;
      key = (s >= minpTh) ? f2key(s) : 0u;
      if (key > x64key) { unsigned p = atomicAdd(&sCnt, 1u); if (p < 64u) sTop[p] = s; }
      s = v.y * invT;
      key = (s >= minpTh) ? f2key(s) : 0u;
      if (key > x64key) { unsigned p = atomicAdd(&sCnt, 1u); if (p < 64u) sTop[p] = s; }
      s = v.z * invT;
      key = (s >= minpTh) ? f2key(s) : 0u;
      if (key > x64key) { unsigned p = atomicAdd(&sCnt, 1u); if (p < 64u) sTop[p] = s; }
      s = v.w * invT;
      key = (s >= minpTh) ? f2key(s) : 0u;
      if (key > x64key) { unsigned p = atomicAdd(&sCnt, 1u); if (p < 64u) sTop[p] = s; }
    }
    __syncthreads();
    if (tid == 0) {
      unsigned c = sCnt;          // = 64 - rank-within-ties, provably <= 63
      if (c > 63u) c = 63u;       // defensive
      float x64v = key2f(x64key);
      for (unsigned j = c; j < 64u; ++j) sTop[j] = x64v;  // tie padding: exact
      // insertion sort, descending (64 elements)
      for (int a = 1; a < 64; ++a) {
        float kv = sTop[a];
        int p = a - 1;
        while (p >= 0 && sTop[p] < kv) { sTop[p + 1] = sTop[p]; --p; }
        sTop[p + 1] = kv;
      }
      // top-k threshold: k-th largest of min-p-masked scaled logits
      int k = top_k[b];
      if (k < 1) k = 1;
      if (k > 64) k = 64;
      float kth = sTop[k - 1];
      // top-p: keep descending position j iff sum of probs above it < p
      float p = top_p[b];
      float Z = sh_Z;
      float D = 0.0f;
      float pth = NEG_INF;
      for (int j = 0; j < 64; ++j) {
        if (!(D < p)) break;
        pth = sTop[j];
        D += expf(sTop[j] - mS) / Z;  // -inf pads contribute 0
      }
      // if top-p set extends past 64, true pth <= sTop[63] <= kth (k<64) => safe
      sh_thresh = fmaxf(kth, pth);
    }
    __syncthreads();
  }

  // ---------------- Pass 7: Gumbel-max argmax over masked logits ------------
  {
    const float thresh = sh_thresh;
    const unsigned rowBase = (unsigned)b * (unsigned)V;
    float bv = NEG_INF;
    int bj = 0x7FFFFFFF;
    for (int i4 = tid; i4 < nvec; i4 += NTHR) {
      float4 v = rowv[i4];
      int base = i4 << 2;
      float s, w, fv, val;
      s = v.x * invT;
      w = (s >= minpTh) ? s : NEG_INF;
      fv = (w >= thresh) ? w : NEG_INF;
      val = fv + gumbel_at(rowBase + (unsigned)(base + 0), half);
      if (val > bv || (val == bv && base + 0 < bj)) { bv = val; bj = base + 0; }
      s = v.y * invT;
      w = (s >= minpTh) ? s : NEG_INF;
      fv = (w >= thresh) ? w : NEG_INF;
      val = fv + gumbel_at(rowBase + (unsigned)(base + 1), half);
      if (val > bv || (val == bv && base + 1 < bj)) { bv = val; bj = base + 1; }
      s = v.z * invT;
      w = (s >= minpTh) ? s : NEG_INF;
      fv = (w >= thresh) ? w : NEG_INF;
      val = fv + gumbel_at(rowBase + (unsigned)(base + 2), half);
      if (val > bv || (val == bv && base + 2 < bj)) { bv = val; bj = base + 2; }
      s = v.w * invT;
      w = (s >= minpTh) ? s : NEG_INF;
      fv = (w >= thresh) ? w : NEG_INF;
      val = fv + gumbel_at(rowBase + (unsigned)(base + 3), half);
      if (val > bv || (val == bv && base + 3 < bj)) { bv = val; bj = base + 3; }
    }
    sV[tid] = bv;
    sI[tid] = bj;
    __syncthreads();
    for (int s = NTHR / 2; s > 0; s >>= 1) {
      if (tid < s) {
        float v2 = sV[tid + s];
        int i2 = sI[tid + s];
        if (v2 > sV[tid] || (v2 == sV[tid] && i2 < sI[tid])) { sV[tid] = v2; sI[tid] = i2; }
      }
      __syncthreads();
    }
    if (tid == 0) {
      float t = temperature[b];
      int pick = (t < 1e-5f) ? sh_greedy : sI[0];
      out[b] = (float)pick;  // indices <= 128000 are exact in fp32
    }
  }
}

extern "C" void kernel_launch(void* const* d_in, const int* in_sizes, int n_in,
                              void* d_out, int out_size, void* d_ws, size_t ws_size,
                              hipStream_t stream) {
  const float* logits = (const float*)d_in[0];
  const float* temperature = (const float*)d_in[1];
  const float* min_p = (const float*)d_in[2];
  const int* top_k = (const int*)d_in[3];
  const float* top_p = (const float*)d_in[4];
  float* out = (float*)d_out;

  const int B = in_sizes[1];                 // 256
  const int V = in_sizes[0] / B;             // 128000 (divisible by 4)
  const unsigned half = (unsigned)(((unsigned long long)B * (unsigned long long)V) >> 1);

  Sampler_28449863369155_kernel<<<dim3(B), dim3(NTHR), 0, stream>>>(
      logits, temperature, min_p, top_k, top_p, out, V, half);
}